// Loss_56410100465732
// MI455X (gfx1250) — compile-verified
//
#include <hip/hip_runtime.h>

typedef __attribute__((ext_vector_type(2))) float v2f;
typedef __attribute__((ext_vector_type(8))) float v8f;

#define KNN      3
#define ALPHA    0.1f
#define NU       1e-3f
#define C_DIM    512
#define N_MEM    16384
#define Q_TOT    16384
#define M_TILE   128          // query rows per workgroup (8 waves x 16 rows)
#define N_CHUNK  16           // memory rows per stage pair, shared by all waves
#define K_STAGE  256          // K columns staged per async stage (2 stages per chunk)
#define XS_STRIDE 514         // 512 + pad -> conflict-free b64 reads (banks 2m)
#define BS_STRIDE 260         // 256 + pad -> banks 4n, rows 16B aligned
#define NBUF     3            // triple buffer -> single barrier per stage

// ws layout (floats): [0]=acc_att [1]=acc_rep [2..3]=pad [4 .. 4+16384)=||m||^2

__global__ void prep_kernel(const float* __restrict__ mb, float* __restrict__ ws) {
  int gwave = (blockIdx.x * blockDim.x + threadIdx.x) >> 5;
  int lane  = threadIdx.x & 31;
  if (blockIdx.x == 0 && threadIdx.x < 2) ws[threadIdx.x] = 0.0f;
  if (gwave < N_MEM) {
    const float* row = mb + (size_t)gwave * C_DIM;
    float s = 0.0f;
    #pragma unroll 4
    for (int c = lane; c < C_DIM; c += 32) { float v = row[c]; s = fmaf(v, v, s); }
    #pragma unroll
    for (int d = 16; d; d >>= 1) s += __shfl_xor(s, d, 32);
    if (lane == 0) ws[4 + gwave] = s;
  }
}

__global__ void __launch_bounds__(256, 1)
dist_topk_kernel(const float* __restrict__ phi, const float* __restrict__ mb,
                 const float* __restrict__ rin, float* __restrict__ ws) {
  extern __shared__ float smem[];
  // Bs first: small LDS offsets -> DS immediate addressing for all B loads
  float* Bs  = smem;                               // [NBUF][N_CHUNK][BS_STRIDE]
  float* x2s = Bs + NBUF * N_CHUNK * BS_STRIDE;    // [M_TILE]
  float* Xs  = x2s + M_TILE;                       // [M_TILE][XS_STRIDE]
  const float* m2 = ws + 4;

  const int tid  = threadIdx.x;
  const int lane = tid & 31;
  const int wave = tid >> 5;
  const int half = lane >> 4;
  const int l16  = lane & 15;

  const unsigned bs_base = (unsigned)(uintptr_t)Bs;   // LDS byte offset of Bs
  const unsigned BUFB    = (unsigned)(N_CHUNK * BS_STRIDE * 4);  // bytes per buffer

  // ---- stage query tile: 128 contiguous hw rows (4096 % 128 == 0) ----
  const int q0  = blockIdx.x * M_TILE;
  const int b   = q0 >> 12;
  const int hw0 = q0 & 4095;
  const float* xbase = phi + (size_t)b * C_DIM * 4096 + hw0;
  for (int idx = tid; idx < M_TILE * C_DIM; idx += 256) {
    int c = idx >> 7;
    int q = idx & 127;
    Xs[q * XS_STRIDE + c] = xbase[(size_t)c * 4096 + q];
  }
  __syncthreads();

  // ---- ||x||^2 per row (2 threads/row; partials staged in Bs scratch) ----
  {
    int row = tid >> 1, h = tid & 1;
    const float* xr = Xs + row * XS_STRIDE + h * 256;
    float s = 0.0f;
    #pragma unroll 8
    for (int c = 0; c < 256; ++c) { float v = xr[c]; s = fmaf(v, v, s); }
    Bs[tid] = s;
  }
  __syncthreads();
  if (tid < M_TILE) x2s[tid] = Bs[2 * tid] + Bs[2 * tid + 1];
  __syncthreads();   // scratch reads done before stage-0 async writes land

  float x2v[8];
  #pragma unroll
  for (int r = 0; r < 8; ++r) x2v[r] = x2s[wave * 16 + r + half * 8];

  const float rv = rin[0];
  const float r2 = rv * rv;

  float t[8][6];
  #pragma unroll
  for (int r = 0; r < 8; ++r)
    #pragma unroll
    for (int j = 0; j < 6; ++j) t[r][j] = __builtin_inff();

  // A fragment source: lane l16 holds query row (wave*16+l16), K pair by half
  const float* arow = Xs + (size_t)(wave * 16 + l16) * XS_STRIDE + 2 * half;

  // ---- async triple-buffered pipeline: stage s = (nb=(s>>1)*16, kc=(s&1)*256) ----
  const int NSTAGE = (N_MEM / N_CHUNK) * 2;

  // one stage: 16 rows x 256 floats = 1024 B128 transfers, 4 per thread
  auto issue_stage = [&](int s, unsigned bufoff) {
    int nb = (s >> 1) * N_CHUNK;
    int kc = (s & 1) * K_STAGE;
    unsigned lbase = bs_base + bufoff;
    #pragma unroll
    for (int i = 0; i < 4; ++i) {
      int f   = tid + 256 * i;
      int row = f >> 6;          // 64 float4 per row
      int c4  = f & 63;
      unsigned goff = (unsigned)((((nb + row) * C_DIM) + kc + c4 * 4) * 4);
      unsigned loff = lbase + (unsigned)((row * BS_STRIDE + c4 * 4) * 4);
      asm volatile("global_load_async_to_lds_b128 %0, %1, %2"
                   :: "v"(loff), "v"(goff), "s"(mb) : "memory");
    }
  };

  issue_stage(0, 0);

  unsigned cons_off = 0;        // buffer byte offset for stage s (consume)
  unsigned prod_off = BUFB;     // buffer byte offset for stage s+1 (produce)
  unsigned cons_idx = 0;        // same in float elements
  float m2n = 0.0f;

  v8f acc = {};
  for (int s = 0; s < NSTAGE; ++s) {
    if (s + 1 < NSTAGE) {
      issue_stage(s + 1, prod_off);
      asm volatile("s_wait_asynccnt 0x4" ::: "memory");   // stage s complete (in-order)
    } else {
      asm volatile("s_wait_asynccnt 0x0" ::: "memory");
    }
    __syncthreads();   // all waves' stage-s data visible; also frees buf (s+2)%3 usage

    if (!(s & 1)) {
      v8f z = {};
      acc = z;                                     // fresh 16-column chunk
      m2n = m2[(s >> 1) * N_CHUNK + l16];          // prefetch ~128 WMMAs ahead of use
    }
    const float* abase = arow + ((s & 1) ? K_STAGE : 0);
    const float* bbase = Bs + cons_idx + (size_t)l16 * BS_STRIDE + 2 * half;
    #pragma unroll 8
    for (int c = 0; c < K_STAGE; c += 4) {
      v2f a  = *(const v2f*)(abase + c);
      v2f bb = *(const v2f*)(bbase + c);
      acc = __builtin_amdgcn_wmma_f32_16x16x4_f32(false, a, false, bb,
                                                  (short)0, acc, false, false);
    }

    if (s & 1) {
      // full K accumulated: d2 = max(||x||^2 + ||m||^2 - 2*dot, 0); top-6 insert
      #pragma unroll
      for (int r = 0; r < 8; ++r) {
        float d2 = fmaxf(fmaf(-2.0f, acc[r], x2v[r] + m2n), 0.0f);
        float a = d2;
        #pragma unroll
        for (int j = 0; j < 6; ++j) {
          float lo = fminf(t[r][j], a);
          float hi = fmaxf(t[r][j], a);
          t[r][j] = lo; a = hi;
        }
      }
    }

    // rotate triple buffers
    cons_off = (cons_off == 2 * BUFB) ? 0u : cons_off + BUFB;
    prod_off = (prod_off == 2 * BUFB) ? 0u : prod_off + BUFB;
    cons_idx = cons_off >> 2;
  }

  // ---- cross-lane (width-16) 6-round pop-merge -> exact per-row top-6 ----
  float att = 0.0f, rep = 0.0f;
  #pragma unroll
  for (int r = 0; r < 8; ++r) {
    float a0 = t[r][0], a1 = t[r][1], a2 = t[r][2];
    float a3 = t[r][3], a4 = t[r][4], a5 = t[r][5];
    float out[6];
    #pragma unroll
    for (int j = 0; j < 6; ++j) {
      float m = a0;
      m = fminf(m, __shfl_xor(m, 1, 16));
      m = fminf(m, __shfl_xor(m, 2, 16));
      m = fminf(m, __shfl_xor(m, 4, 16));
      m = fminf(m, __shfl_xor(m, 8, 16));
      out[j] = m;
      bool pop = (a0 == m);
      a0 = pop ? a1 : a0; a1 = pop ? a2 : a1; a2 = pop ? a3 : a2;
      a3 = pop ? a4 : a3; a4 = pop ? a5 : a4;
      a5 = pop ? __builtin_inff() : a5;
    }
    if (l16 == 0) {  // lanes 0 & 16 own rows r and r+8 of this wave's subtile
      att += fmaxf(out[0] - r2, 0.0f) + fmaxf(out[1] - r2, 0.0f) + fmaxf(out[2] - r2, 0.0f);
      rep += fmaxf(r2 - out[3] - ALPHA, 0.0f) + fmaxf(r2 - out[4] - ALPHA, 0.0f)
           + fmaxf(r2 - out[5] - ALPHA, 0.0f);
    }
  }
  if (l16 == 0) {
    atomicAdd(&ws[0], att);
    atomicAdd(&ws[1], rep);
  }
}

__global__ void finalize_kernel(const float* __restrict__ ws, float* __restrict__ out) {
  out[0] = (ws[0] + ws[1]) * (1.0f / ((float)(Q_TOT * KNN) * NU));
}

extern "C" void kernel_launch(void* const* d_in, const int* in_sizes, int n_in,
                              void* d_out, int out_size, void* d_ws, size_t ws_size,
                              hipStream_t stream) {
  const float* phi = (const float*)d_in[0];
  const float* mb  = (const float*)d_in[1];
  const float* rin = (const float*)d_in[2];
  float* out = (float*)d_out;
  float* ws  = (float*)d_ws;

  prep_kernel<<<(N_MEM * 32) / 256, 256, 0, stream>>>(mb, ws);

  size_t shmem = (size_t)(NBUF * N_CHUNK * BS_STRIDE + M_TILE + M_TILE * XS_STRIDE) * sizeof(float);
  hipFuncSetAttribute(reinterpret_cast<const void*>(dist_topk_kernel),
                      hipFuncAttributeMaxDynamicSharedMemorySize, (int)shmem);
  dist_topk_kernel<<<Q_TOT / M_TILE, 256, shmem, stream>>>(phi, mb, rin, ws);

  finalize_kernel<<<1, 1, 0, stream>>>(ws, out);
}